// BarycentricCoordinates_29291676959134
// MI455X (gfx1250) — compile-verified
//
#include <hip/hip_runtime.h>
#include <math.h>

#define N_PTS   8192
#define KCAP    96
#define RADIUS  0.27f
#define RAD2    (0.27f * 0.27f)
#define TRADIUS 0.2f
#define NRAD    5
#define NANG    8
#define NCELL   (NRAD * NANG)

typedef float v2f __attribute__((ext_vector_type(2)));
typedef float v8f __attribute__((ext_vector_type(8)));

// ---------------------------------------------------------------------------
// Kernel 0: zero the per-point neighbor counters
// ---------------------------------------------------------------------------
__global__ void k_zero_cnt(int* __restrict__ cnt) {
    int i = blockIdx.x * blockDim.x + threadIdx.x;
    if (i < N_PTS) cnt[i] = 0;
}

// ---------------------------------------------------------------------------
// Kernel 1: fused Gram-tile (WMMA f32 16x16x4) -> distance -> radius append.
// One block = one 16-row tile; 8 waves sweep the 512 column tiles.
// ---------------------------------------------------------------------------
__global__ __launch_bounds__(256) void k_neighbors(const float* __restrict__ X,
                                                   int* __restrict__ nbr_idx,
                                                   int* __restrict__ nbr_cnt) {
    __shared__ int   s_cnt[16];
    __shared__ float s_rsq[16];

    const int tid     = threadIdx.x;
    const int lane    = tid & 31;
    const int wave    = tid >> 5;        // 0..7
    const int rowBase = blockIdx.x * 16;

    if (tid < 16) {
        float px = X[(rowBase + tid) * 3 + 0];
        float py = X[(rowBase + tid) * 3 + 1];
        float pz = X[(rowBase + tid) * 3 + 2];
        s_rsq[tid] = px * px + py * py + pz * pz;
        s_cnt[tid] = 0;
    }
    __syncthreads();

    // A operand: 16x4 row tile, K=3 padded to 4 with zero.
    // lanes 0-15: VGPR0=K0, VGPR1=K1 ; lanes 16-31: VGPR0=K2, VGPR1=K3(=0)
    v2f a;
    {
        int m = rowBase + (lane & 15);
        float mx = X[m * 3 + 0], my = X[m * 3 + 1], mz = X[m * 3 + 2];
        if (lane < 16) { a.x = mx; a.y = my; }
        else           { a.x = mz; a.y = 0.0f; }
    }

    for (int jt = wave; jt < N_PTS / 16; jt += 8) {
        const int colBase = jt * 16;
        const int n = colBase + (lane & 15);
        float nx = X[n * 3 + 0], ny = X[n * 3 + 1], nz = X[n * 3 + 2];

        // B operand: 4x16, rows striped like C/D (VGPR v holds rows v, v+2)
        v2f b;
        if (lane < 16) { b.x = nx; b.y = ny; }   // K0, K1
        else           { b.x = nz; b.y = 0.0f; } // K2, K3(=0)

        v8f c = {};
        c = __builtin_amdgcn_wmma_f32_16x16x4_f32(
                /*neg_a=*/false, a, /*neg_b=*/false, b,
                /*c_mod=*/(short)0, c, /*reuse_a=*/false, /*reuse_b=*/false);

        const float nsq  = nx * nx + ny * ny + nz * nz;
        const int   mOff = (lane < 16) ? 0 : 8;
#pragma unroll
        for (int v = 0; v < 8; ++v) {
            int   mloc = v + mOff;
            float d2   = s_rsq[mloc] + nsq - 2.0f * c[v];
            if (d2 <= RAD2) {
                int slot = atomicAdd(&s_cnt[mloc], 1);   // ds_add_rtn_u32
                if (slot < KCAP)
                    nbr_idx[(rowBase + mloc) * KCAP + slot] = n;
            }
        }
    }
    __syncthreads();
    if (tid < 16) nbr_cnt[rowBase + tid] = min(s_cnt[tid], KCAP);
}

// ---------------------------------------------------------------------------
// 3x3 symmetric eigen helpers
// ---------------------------------------------------------------------------
__device__ inline void evec_from_lambda(float a00, float a01, float a02,
                                        float a11, float a12, float a22,
                                        float lam, float fbx, float fby, float fbz,
                                        float ev[3]) {
    float r0x = a00 - lam, r0y = a01,       r0z = a02;
    float r1x = a01,       r1y = a11 - lam, r1z = a12;
    float r2x = a02,       r2y = a12,       r2z = a22 - lam;

    float c0x = r0y * r1z - r0z * r1y, c0y = r0z * r1x - r0x * r1z, c0z = r0x * r1y - r0y * r1x;
    float c1x = r0y * r2z - r0z * r2y, c1y = r0z * r2x - r0x * r2z, c1z = r0x * r2y - r0y * r2x;
    float c2x = r1y * r2z - r1z * r2y, c2y = r1z * r2x - r1x * r2z, c2z = r1x * r2y - r1y * r2x;

    float n0 = c0x * c0x + c0y * c0y + c0z * c0z;
    float n1 = c1x * c1x + c1y * c1y + c1z * c1z;
    float n2 = c2x * c2x + c2y * c2y + c2z * c2z;

    float bx = c0x, by = c0y, bz = c0z, bn = n0;
    if (n1 > bn) { bx = c1x; by = c1y; bz = c1z; bn = n1; }
    if (n2 > bn) { bx = c2x; by = c2y; bz = c2z; bn = n2; }

    if (bn > 1e-30f) {
        float inv = rsqrtf(bn);
        ev[0] = bx * inv; ev[1] = by * inv; ev[2] = bz * inv;
    } else {
        ev[0] = fbx; ev[1] = fby; ev[2] = fbz;   // degenerate fallback
    }
}

// ---------------------------------------------------------------------------
// Kernel 2: SHOT local reference frame per point
// ---------------------------------------------------------------------------
__global__ void k_lrf(const float* __restrict__ X,
                      const int* __restrict__ nbr_idx,
                      const int* __restrict__ nbr_cnt,
                      float* __restrict__ lrf) {
    int i = blockIdx.x * blockDim.x + threadIdx.x;
    if (i >= N_PTS) return;

    const int cnt = nbr_cnt[i];
    const float cx = X[i * 3 + 0], cy = X[i * 3 + 1], cz = X[i * 3 + 2];

    float c00 = 0.f, c01 = 0.f, c02 = 0.f, c11 = 0.f, c12 = 0.f, c22 = 0.f, wsum = 0.f;
    for (int k = 0; k < cnt; ++k) {
        int j = nbr_idx[i * KCAP + k];
        float vx = X[j * 3 + 0] - cx, vy = X[j * 3 + 1] - cy, vz = X[j * 3 + 2] - cz;
        float d  = sqrtf(vx * vx + vy * vy + vz * vz);
        float w  = fabsf(RADIUS - d);
        c00 += w * vx * vx; c01 += w * vx * vy; c02 += w * vx * vz;
        c11 += w * vy * vy; c12 += w * vy * vz; c22 += w * vz * vz;
        wsum += w;
    }
    float inv = (wsum > 1e-30f) ? (1.0f / wsum) : 0.0f;
    c00 *= inv; c01 *= inv; c02 *= inv; c11 *= inv; c12 *= inv; c22 *= inv;

    // analytic eigenvalues (lam1 >= lam2 >= lam3)
    float q  = (c00 + c11 + c22) * (1.0f / 3.0f);
    float p1 = c01 * c01 + c02 * c02 + c12 * c12;
    float b00 = c00 - q, b11 = c11 - q, b22 = c22 - q;
    float p2 = b00 * b00 + b11 * b11 + b22 * b22 + 2.0f * p1;
    float p  = sqrtf(p2 * (1.0f / 6.0f));

    float lam1, lam3;
    if (p < 1e-20f) {
        lam1 = q; lam3 = q;
    } else {
        float ip = 1.0f / p;
        float detB = (b00 * (b11 * b22 - c12 * c12)
                    - c01 * (c01 * b22 - c12 * c02)
                    + c02 * (c01 * c12 - b11 * c02)) * ip * ip * ip;
        float rr  = fminf(fmaxf(0.5f * detB, -1.0f), 1.0f);
        float phi = acosf(rr) * (1.0f / 3.0f);
        lam1 = q + 2.0f * p * cosf(phi);
        lam3 = q + 2.0f * p * cosf(phi + 2.0943951023931953f);
    }

    float xax[3], zax[3];
    evec_from_lambda(c00, c01, c02, c11, c12, c22, lam1, 1.f, 0.f, 0.f, xax);
    evec_from_lambda(c00, c01, c02, c11, c12, c22, lam3, 0.f, 0.f, 1.f, zax);

    // sign disambiguation against real neighbors
    int posx = 0, negx = 0, posz = 0, negz = 0;
    for (int k = 0; k < cnt; ++k) {
        int j = nbr_idx[i * KCAP + k];
        float vx = X[j * 3 + 0] - cx, vy = X[j * 3 + 1] - cy, vz = X[j * 3 + 2] - cz;
        float dx = vx * xax[0] + vy * xax[1] + vz * xax[2];
        float dz = vx * zax[0] + vy * zax[1] + vz * zax[2];
        if (dx >= 0.0f) posx++; else negx++;
        if (dz >= 0.0f) posz++; else negz++;
    }
    if (posx < negx) { xax[0] = -xax[0]; xax[1] = -xax[1]; xax[2] = -xax[2]; }
    if (posz < negz) { zax[0] = -zax[0]; zax[1] = -zax[1]; zax[2] = -zax[2]; }

    float yax[3] = { zax[1] * xax[2] - zax[2] * xax[1],
                     zax[2] * xax[0] - zax[0] * xax[2],
                     zax[0] * xax[1] - zax[1] * xax[0] };

    float* o = lrf + (size_t)i * 9;
    o[0] = zax[0]; o[1] = zax[1]; o[2] = zax[2];
    o[3] = yax[0]; o[4] = yax[1]; o[5] = yax[2];
    o[6] = xax[0]; o[7] = xax[1]; o[8] = xax[2];
}

// ---------------------------------------------------------------------------
// Kernel 3: tangent-plane projection per (point, neighbor)
// ---------------------------------------------------------------------------
__global__ void k_proj(const float* __restrict__ X,
                       const int* __restrict__ nbr_idx,
                       const int* __restrict__ nbr_cnt,
                       const float* __restrict__ lrf,
                       float* __restrict__ proj) {
    int g = blockIdx.x * blockDim.x + threadIdx.x;
    if (g >= N_PTS * KCAP) return;
    int i = g / KCAP, k = g - i * KCAP;
    if (k >= nbr_cnt[i]) return;

    const float* L = lrf + (size_t)i * 9;
    float zx = L[0], zy = L[1], zz = L[2];
    float yx = L[3], yy = L[4], yz = L[5];
    float xx = L[6], xy = L[7], xz = L[8];

    int j = nbr_idx[i * KCAP + k];
    float vx = X[j * 3 + 0] - X[i * 3 + 0];
    float vy = X[j * 3 + 1] - X[i * 3 + 1];
    float vz = X[j * 3 + 2] - X[i * 3 + 2];

    float dz = vx * zx + vy * zy + vz * zz;
    float tx = vx - dz * zx, ty = vy - dz * zy, tz = vz - dz * zz;

    proj[(size_t)g * 2 + 0] = tx * yx + ty * yy + tz * yz;   // basis row 1 (y)
    proj[(size_t)g * 2 + 1] = tx * xx + ty * xy + tz * xz;   // basis row 2 (x)
}

// ---------------------------------------------------------------------------
// Kernel 4: 3-nearest template match + barycentric weights per (point, cell)
// ---------------------------------------------------------------------------
__global__ void k_bary(const float* __restrict__ proj,
                       const int* __restrict__ nbr_idx,
                       const int* __restrict__ nbr_cnt,
                       float* __restrict__ out) {
    int g = blockIdx.x * blockDim.x + threadIdx.x;
    if (g >= N_PTS * NCELL) return;
    int n = g / NCELL, cell = g - n * NCELL;
    int r = cell / NANG, a = cell - r * NANG;

    float rad = ((float)(r + 1) / (float)NRAD) * TRADIUS;
    float th  = (float)a * (6.283185307179586f / (float)NANG);
    float tx  = rad * cosf(th);
    float ty  = rad * sinf(th);

    int cnt = nbr_cnt[n];
    float b0 = 3.4e38f, b1 = 3.4e38f, b2 = 3.4e38f;
    int   i0 = 0, i1 = 0, i2 = 0;
    for (int k = 0; k < cnt; ++k) {
        float px = proj[((size_t)n * KCAP + k) * 2 + 0];
        float py = proj[((size_t)n * KCAP + k) * 2 + 1];
        float dx = px - tx, dy = py - ty;
        float d  = dx * dx + dy * dy;
        if (d < b0)      { b2 = b1; i2 = i1; b1 = b0; i1 = i0; b0 = d; i0 = k; }
        else if (d < b1) { b2 = b1; i2 = i1; b1 = d;  i1 = k; }
        else if (d < b2) { b2 = d;  i2 = k; }
    }

    size_t base = ((size_t)n * NCELL + cell) * 6;
    if (cnt >= 3) {
        float c0x = proj[((size_t)n * KCAP + i0) * 2 + 0], c0y = proj[((size_t)n * KCAP + i0) * 2 + 1];
        float c1x = proj[((size_t)n * KCAP + i1) * 2 + 0], c1y = proj[((size_t)n * KCAP + i1) * 2 + 1];
        float c2x = proj[((size_t)n * KCAP + i2) * 2 + 0], c2y = proj[((size_t)n * KCAP + i2) * 2 + 1];

        float v0x = c2x - c0x, v0y = c2y - c0y;
        float v1x = c1x - c0x, v1y = c1y - c0y;
        float v2x = tx  - c0x, v2y = ty  - c0y;

        float d00 = v0x * v0x + v0y * v0y;
        float d01 = v0x * v1x + v0y * v1y;
        float d02 = v0x * v2x + v0y * v2y;
        float d11 = v1x * v1x + v1y * v1y;
        float d12 = v1x * v2x + v1y * v2y;
        float den = d00 * d11 - d01 * d01;
        float w2  = (d11 * d02 - d01 * d12) / den;
        float w1  = (d00 * d12 - d01 * d02) / den;
        float w0  = 1.0f - w2 - w1;

        out[base + 0] = (float)nbr_idx[n * KCAP + i0]; out[base + 1] = w0;
        out[base + 2] = (float)nbr_idx[n * KCAP + i1]; out[base + 3] = w1;
        out[base + 4] = (float)nbr_idx[n * KCAP + i2]; out[base + 5] = w2;
    } else {
        // reference: idx3 forced to 0 -> gidx = nbr_idx[n][0] = self (= n); bc = 0
        float gi = (float)n;
        out[base + 0] = gi; out[base + 1] = 0.0f;
        out[base + 2] = gi; out[base + 3] = 0.0f;
        out[base + 4] = gi; out[base + 5] = 0.0f;
    }
}

// ---------------------------------------------------------------------------
// Host-side launch
// ---------------------------------------------------------------------------
extern "C" void kernel_launch(void* const* d_in, const int* in_sizes, int n_in,
                              void* d_out, int out_size, void* d_ws, size_t ws_size,
                              hipStream_t stream) {
    (void)in_sizes; (void)n_in; (void)out_size; (void)ws_size;

    const float* X = (const float*)d_in[0];       // (8192, 3) float32
    float* out     = (float*)d_out;               // (8192, 5, 8, 3, 2) float32

    char* ws = (char*)d_ws;
    int*   nbr_cnt = (int*)(ws);                                  //  32 KB
    int*   nbr_idx = (int*)(ws + 32768);                          // 3.0 MB
    float* lrf     = (float*)(ws + 32768 + 3145728);              // 288 KB
    float* proj    = (float*)(ws + 32768 + 3145728 + 294912);     // 6.0 MB

    k_zero_cnt<<<N_PTS / 256, 256, 0, stream>>>(nbr_cnt);
    k_neighbors<<<N_PTS / 16, 256, 0, stream>>>(X, nbr_idx, nbr_cnt);
    k_lrf<<<N_PTS / 256, 256, 0, stream>>>(X, nbr_idx, nbr_cnt, lrf);
    k_proj<<<(N_PTS * KCAP) / 256, 256, 0, stream>>>(X, nbr_idx, nbr_cnt, lrf, proj);
    k_bary<<<(N_PTS * NCELL) / 256, 256, 0, stream>>>(proj, nbr_idx, nbr_cnt, out);
}